// TokenNetwork_3367254360562
// MI455X (gfx1250) — compile-verified
//
#include <hip/hip_runtime.h>

// ---------------------------------------------------------------------------
// 3-layer GCN (PyG GCNConv, no self-loops) + global_add_pool on MI455X.
//   N=50000 nodes, E=800000 edges, F=HID=128, 256 graphs.
//
// Strategy (gfx1250, wave32):
//   * GEMM h = x @ W via V_WMMA_F32_16X16X4_F32 (full fp32 precision).
//   * Build CSR (edges grouped by destination) ONCE, then each layer is a
//     pure gather-reduce: wave-per-node register accumulation with fused
//     bias+ReLU. This replaces 102M f32 atomics/layer with coalesced 512B
//     loads; h (25.6MB) and the CSR arrays are L2-resident (192MB L2).
//   * Readout exploits sorted `batch`: block-per-graph binary search +
//     register/LDS reduction, zero atomics.
// ---------------------------------------------------------------------------

#define NNODES 50000
#define NEDGES 800000
#define FDIM   128
#define NGRAPH 256

typedef float v2f __attribute__((ext_vector_type(2)));
typedef float v8f __attribute__((ext_vector_type(8)));

__device__ __forceinline__ void fadd(float* p, float v) {
  // relaxed agent-scope fadd -> global_atomic_add_f32 (no return, STOREcnt)
  __hip_atomic_fetch_add(p, v, __ATOMIC_RELAXED, __HIP_MEMORY_SCOPE_AGENT);
}

// ---------------------------------------------------------------------------
__global__ void k_zero(float4* __restrict__ p, int n4) {
  int t = blockIdx.x * blockDim.x + threadIdx.x;
  if (t < n4) p[t] = make_float4(0.f, 0.f, 0.f, 0.f);
}

// deg[i] = weighted in-degree (sum of w over edges with dst==i)
// cnt[i] = integer in-degree (for CSR build)
__global__ void k_deg(const int* __restrict__ col, const float* __restrict__ w,
                      float* __restrict__ deg, int* __restrict__ cnt) {
  int e = blockIdx.x * blockDim.x + threadIdx.x;
  if (e >= NEDGES) return;
  int d = col[e];
  fadd(&deg[d], w[e]);
  atomicAdd(&cnt[d], 1);
}

// Single-block exclusive scan over cnt[0..N) -> ptr (and cursor copy).
__global__ void k_scan(const int* __restrict__ cnt, int* __restrict__ ptr,
                       int* __restrict__ cursor) {
  __shared__ int sums[256];
  const int CH = (NNODES + 255) / 256;  // 196 elements per thread
  const int t = threadIdx.x;
  int s = 0;
  for (int j = 0; j < CH; ++j) {
    int i = t * CH + j;
    if (i < NNODES) s += cnt[i];
  }
  sums[t] = s;
  __syncthreads();
  if (t == 0) {
    int run = 0;
    for (int k = 0; k < 256; ++k) { int v = sums[k]; sums[k] = run; run += v; }
    ptr[NNODES] = run;  // == NEDGES
  }
  __syncthreads();
  int run = sums[t];
  for (int j = 0; j < CH; ++j) {
    int i = t * CH + j;
    if (i < NNODES) { ptr[i] = run; cursor[i] = run; run += cnt[i]; }
  }
}

// Scatter edges into destination-sorted CSR order; compute norm_e inline.
__global__ void k_fill(const int* __restrict__ row, const int* __restrict__ col,
                       const float* __restrict__ w, const float* __restrict__ deg,
                       int* __restrict__ cursor, int* __restrict__ srcs,
                       float* __restrict__ wvals) {
  int e = blockIdx.x * blockDim.x + threadIdx.x;
  if (e >= NEDGES) return;
  int r = row[e], d = col[e];
  int p = atomicAdd(&cursor[d], 1);
  float dr = deg[r], dc = deg[d];
  float ir = dr > 0.f ? rsqrtf(dr) : 0.f;
  float ic = dc > 0.f ? rsqrtf(dc) : 0.f;
  srcs[p] = r;
  wvals[p] = ir * w[e] * ic;
}

// ---------------------------------------------------------------------------
// H[N,128] = X[N,128] @ W[128,128]   (fp32 WMMA, 16x16x4, wave32)
// Block = 256 threads = 8 waves; block owns 16 rows, each wave owns 16 cols.
// ---------------------------------------------------------------------------
#define XS_STRIDE 132  // pad 128->132: conflict-free column reads across lanes

__global__ __launch_bounds__(256)
void k_gemm(const float* __restrict__ X, const float* __restrict__ Wg,
            float* __restrict__ H) {
  __shared__ float Xs[16 * XS_STRIDE];
  const int tid = threadIdx.x;
  const int rowBase = blockIdx.x * 16;  // 50000 % 16 == 0, no guard needed

  const float4* Xg4 = (const float4*)(X + (size_t)rowBase * FDIM);
#pragma unroll
  for (int i = 0; i < 2; ++i) {
    int idx = tid + i * 256;      // 0..511
    int r = idx >> 5;             // row 0..15
    int c4 = idx & 31;            // float4 col 0..31
    float4 v = Xg4[r * 32 + c4];
    int o = r * XS_STRIDE + c4 * 4;
    Xs[o + 0] = v.x; Xs[o + 1] = v.y; Xs[o + 2] = v.z; Xs[o + 3] = v.w;
  }
  __syncthreads();

  const int wave = tid >> 5;
  const int lane = tid & 31;
  const int m    = lane & 15;          // A row / output column within tile
  const int hi   = lane >> 4;          // half-wave: K pair select
  const int koff = hi * 2;             // lanes 0-15: K{0,1}; 16-31: K{2,3}
  const int n    = (wave << 4) + m;    // output column (per-wave 16-col slab)

  v8f c = {};
#pragma unroll 8
  for (int k0 = 0; k0 < FDIM; k0 += 4) {
    int k = k0 + koff;
    v2f a, b;
    a.x = Xs[m * XS_STRIDE + k];
    a.y = Xs[m * XS_STRIDE + k + 1];
    b.x = Wg[k * FDIM + n];
    b.y = Wg[(k + 1) * FDIM + n];
    c = __builtin_amdgcn_wmma_f32_16x16x4_f32(
        false, a, false, b, (short)0, c, false, false);
  }

#pragma unroll
  for (int vr = 0; vr < 8; ++vr) {
    int rl = vr + (hi << 3);
    H[(size_t)(rowBase + rl) * FDIM + n] = c[vr];
  }
}

// ---------------------------------------------------------------------------
// out[i,:] = act( sum_{p in [ptr[i],ptr[i+1])} wvals[p] * H[srcs[p],:] + b )
// One wave per node; float4 per lane covers 128 features; 2-edge unroll.
// ---------------------------------------------------------------------------
template <bool RELU>
__global__ __launch_bounds__(256)
void k_gather(const int* __restrict__ ptr, const int* __restrict__ srcs,
              const float* __restrict__ wvals, const float4* __restrict__ H4,
              const float4* __restrict__ bias4, float4* __restrict__ out4) {
  const int wave = threadIdx.x >> 5;
  const int lane = threadIdx.x & 31;
  const int i = blockIdx.x * 8 + wave;  // node; 50000 % 8 == 0

  int p = ptr[i], pend = ptr[i + 1];
  float4 acc = make_float4(0.f, 0.f, 0.f, 0.f);
  for (; p + 1 < pend; p += 2) {
    int   s0 = srcs[p],   s1 = srcs[p + 1];
    float w0 = wvals[p],  w1 = wvals[p + 1];
    float4 v0 = H4[(size_t)s0 * 32 + lane];
    float4 v1 = H4[(size_t)s1 * 32 + lane];
    acc.x = fmaf(w0, v0.x, acc.x); acc.y = fmaf(w0, v0.y, acc.y);
    acc.z = fmaf(w0, v0.z, acc.z); acc.w = fmaf(w0, v0.w, acc.w);
    acc.x = fmaf(w1, v1.x, acc.x); acc.y = fmaf(w1, v1.y, acc.y);
    acc.z = fmaf(w1, v1.z, acc.z); acc.w = fmaf(w1, v1.w, acc.w);
  }
  if (p < pend) {
    int s0 = srcs[p]; float w0 = wvals[p];
    float4 v0 = H4[(size_t)s0 * 32 + lane];
    acc.x = fmaf(w0, v0.x, acc.x); acc.y = fmaf(w0, v0.y, acc.y);
    acc.z = fmaf(w0, v0.z, acc.z); acc.w = fmaf(w0, v0.w, acc.w);
  }
  float4 b = bias4[lane];
  acc.x += b.x; acc.y += b.y; acc.z += b.z; acc.w += b.w;
  if (RELU) {
    acc.x = fmaxf(acc.x, 0.f); acc.y = fmaxf(acc.y, 0.f);
    acc.z = fmaxf(acc.z, 0.f); acc.w = fmaxf(acc.w, 0.f);
  }
  out4[(size_t)i * 32 + lane] = acc;
}

// ---------------------------------------------------------------------------
// global_add_pool: batch is sorted, so block g binary-searches its node
// range and reduces with registers + LDS. Zero atomics.
// ---------------------------------------------------------------------------
__device__ __forceinline__ int lowerBound(const int* __restrict__ a, int n, int key) {
  int lo = 0, hi = n;
  while (lo < hi) { int mid = (lo + hi) >> 1; if (a[mid] < key) lo = mid + 1; else hi = mid; }
  return lo;
}

__global__ __launch_bounds__(256)
void k_readout(const float4* __restrict__ h4, const int* __restrict__ batch,
               float4* __restrict__ ro4) {
  __shared__ float4 red[8 * 32];
  const int g = blockIdx.x;
  const int wave = threadIdx.x >> 5;
  const int lane = threadIdx.x & 31;
  const int lo = lowerBound(batch, NNODES, g);
  const int hi = lowerBound(batch, NNODES, g + 1);

  float4 acc = make_float4(0.f, 0.f, 0.f, 0.f);
  for (int i = lo + wave; i < hi; i += 8) {
    float4 v = h4[(size_t)i * 32 + lane];
    acc.x += v.x; acc.y += v.y; acc.z += v.z; acc.w += v.w;
  }
  red[wave * 32 + lane] = acc;
  __syncthreads();
  if (wave == 0) {
#pragma unroll
    for (int w = 1; w < 8; ++w) {
      float4 v = red[w * 32 + lane];
      acc.x += v.x; acc.y += v.y; acc.z += v.z; acc.w += v.w;
    }
    ro4[(size_t)g * 32 + lane] = acc;
  }
}

// ---------------------------------------------------------------------------
extern "C" void kernel_launch(void* const* d_in, const int* in_sizes, int n_in,
                              void* d_out, int out_size, void* d_ws, size_t ws_size,
                              hipStream_t stream) {
  const float* gx    = (const float*)d_in[0];
  const int*   eidx  = (const int*)d_in[1];   // [2, E]
  const int*   batch = (const int*)d_in[2];   // sorted
  const float* ew    = (const float*)d_in[3];
  const float* W1 = (const float*)d_in[4]; const float* b1 = (const float*)d_in[5];
  const float* W2 = (const float*)d_in[6]; const float* b2 = (const float*)d_in[7];
  const float* W3 = (const float*)d_in[8]; const float* b3 = (const float*)d_in[9];
  const int* row = eidx;
  const int* col = eidx + NEDGES;

  // workspace carve-out (256B aligned): ~58.5 MB total
  char* ws = (char*)d_ws;
  size_t off = 0;
  auto carve = [&](size_t bytes) -> void* {
    void* p = ws + off;
    off = (off + bytes + 255) & ~(size_t)255;
    return p;
  };
  float* deg    = (float*)carve((size_t)NNODES * 4);
  int*   cnt    = (int*)  carve((size_t)NNODES * 4);
  int*   ptr    = (int*)  carve((size_t)(NNODES + 1) * 4);
  int*   cursor = (int*)  carve((size_t)NNODES * 4);
  int*   srcs   = (int*)  carve((size_t)NEDGES * 4);
  float* wvals  = (float*)carve((size_t)NEDGES * 4);
  float* htmp   = (float*)carve((size_t)NNODES * FDIM * 4);
  float* xalt   = (float*)carve((size_t)NNODES * FDIM * 4);

  float* h_out = (float*)d_out;                    // [N, 128]
  float* ro    = h_out + (size_t)NNODES * FDIM;    // [256, 128]

  const int NT = 256;

  // --- CSR build (once; shared by all three layers) ---
  k_zero<<<(NNODES / 4 + NT - 1) / NT, NT, 0, stream>>>((float4*)deg, NNODES / 4);
  k_zero<<<(NNODES / 4 + NT - 1) / NT, NT, 0, stream>>>((float4*)cnt, NNODES / 4);
  k_deg <<<NEDGES / NT, NT, 0, stream>>>(col, ew, deg, cnt);
  k_scan<<<1, NT, 0, stream>>>(cnt, ptr, cursor);
  k_fill<<<NEDGES / NT, NT, 0, stream>>>(row, col, ew, deg, cursor, srcs, wvals);

  // --- 3 GCN layers: WMMA GEMM + fused gather/bias/act ---
  auto layer = [&](const float* xin, const float* Wl, const float* bl,
                   float* xout, bool relu) {
    k_gemm<<<NNODES / 16, NT, 0, stream>>>(xin, Wl, htmp);
    if (relu)
      k_gather<true ><<<NNODES / 8, NT, 0, stream>>>(
          ptr, srcs, wvals, (const float4*)htmp, (const float4*)bl, (float4*)xout);
    else
      k_gather<false><<<NNODES / 8, NT, 0, stream>>>(
          ptr, srcs, wvals, (const float4*)htmp, (const float4*)bl, (float4*)xout);
  };

  layer(gx,   W1, b1, xalt,  true);
  layer(xalt, W2, b2, xalt,  true);
  layer(xalt, W3, b3, h_out, false);

  // --- readout (block-per-graph, no atomics) ---
  k_readout<<<NGRAPH, NT, 0, stream>>>((const float4*)h_out, batch, (float4*)ro);
}